// NodeAttnModel_21492016349616
// MI455X (gfx1250) — compile-verified
//
#include <hip/hip_runtime.h>
#include <hip/hip_bf16.h>

#define NN 50000
#define EE 500000
#define DD 160
#define HH 5
#define CC 32
#define MH 50
#define EPSLN 1e-5f
#define NEG_SLOPE 0.2f

typedef float v2f __attribute__((ext_vector_type(2)));
typedef float v8f __attribute__((ext_vector_type(8)));

__device__ __forceinline__ v8f wmma_f32(v2f a, v2f b, v8f c) {
    // D = A(16x4 f32) * B(4x16 f32) + C(16x16 f32)
    return __builtin_amdgcn_wmma_f32_16x16x4_f32(
        /*neg_a=*/false, a, /*neg_b=*/false, b,
        /*c_mod=*/(short)0, c, /*reuse_a=*/false, /*reuse_b=*/false);
}

__device__ __forceinline__ float wave_sum(float v) {
#pragma unroll
    for (int m = 1; m < 32; m <<= 1) v += __shfl_xor(v, m, 32);
    return v;
}

// order-preserving float -> uint key for atomicMax-based segment max
__device__ __forceinline__ unsigned fkey(float f) {
    unsigned b = __float_as_uint(f);
    return (b & 0x80000000u) ? ~b : (b | 0x80000000u);
}
__device__ __forceinline__ float fdecode(unsigned u) {
    unsigned b = (u & 0x80000000u) ? (u & 0x7FFFFFFFu) : ~u;
    return __uint_as_float(b);
}

// ---------------------------------------------------------------------------
// Kernel 1: xl = x@Wl + bl, xr = x@Wr + br.  One wave per 16x16 output tile,
// both weight matrices at once (shared A loads).  K = 160 -> 40 WMMAs each.
// ---------------------------------------------------------------------------
__global__ __launch_bounds__(256) void node_xform(
    const float* __restrict__ x,
    const float* __restrict__ Wl, const float* __restrict__ bl,
    const float* __restrict__ Wr, const float* __restrict__ br,
    float* __restrict__ xl, float* __restrict__ xr)
{
    const int wid  = (blockIdx.x * blockDim.x + threadIdx.x) >> 5;
    const int lane = threadIdx.x & 31;
    const int nt = wid % (DD / 16);   // output-column tile
    const int mt = wid / (DD / 16);   // row tile
    if (mt >= NN / 16) return;

    const int hl  = lane >> 4;        // K-pair half (0/1)
    const int l16 = lane & 15;
    const int arow = mt * 16 + l16;   // A row handled by this lane
    const int coln = nt * 16 + l16;   // output col handled by this lane

    const float* __restrict__ ap = x + (size_t)arow * DD + 2 * hl;

    v8f cl = {}; v8f cr = {};
#pragma unroll 4
    for (int kb = 0; kb < DD / 4; ++kb) {
        const int k2 = 4 * kb + 2 * hl;
        v2f a = *(const v2f*)(ap + 4 * kb);
        v2f bL, bR;
        bL[0] = Wl[(size_t)k2 * DD + coln];
        bL[1] = Wl[(size_t)(k2 + 1) * DD + coln];
        bR[0] = Wr[(size_t)k2 * DD + coln];
        bR[1] = Wr[(size_t)(k2 + 1) * DD + coln];
        cl = wmma_f32(a, bL, cl);
        cr = wmma_f32(a, bR, cr);
    }

    const int rbase = mt * 16 + hl * 8;
    const float addl = bl[coln], addr_ = br[coln];
#pragma unroll
    for (int v = 0; v < 8; ++v) {
        const size_t o = (size_t)(rbase + v) * DD + coln;
        xl[o] = cl[v] + addl;
        xr[o] = cr[v] + addr_;
    }
}

// ---------------------------------------------------------------------------
// Kernel 2: fused edge scores.  One wave per 16-edge tile.  The 16x160 A tile
// (edge_attr rows) is preloaded ONCE into 40 v2f register fragments and reused
// across all 10 column sub-tiles (5 heads x 2).  Per sub-tile: 40 WMMAs for
// the edge_attr@We tile, add gathered xl[src] / xr[dst], leaky-relu, scale by
// att, butterfly-reduce over 16 columns -> score[E,5].  ea never hits memory.
// ---------------------------------------------------------------------------
__global__ __launch_bounds__(256) void edge_score(
    const float* __restrict__ edge_attr, const float* __restrict__ We,
    const int* __restrict__ ei,
    const float* __restrict__ xl, const float* __restrict__ xr,
    const float* __restrict__ att, float* __restrict__ score)
{
    const int wid  = (blockIdx.x * blockDim.x + threadIdx.x) >> 5;
    const int lane = threadIdx.x & 31;
    if (wid >= EE / 16) return;

    const int hl  = lane >> 4;
    const int l16 = lane & 15;
    const int rbase = wid * 16 + hl * 8;   // first edge row of this lane's half

    int sidx[8], didx[8];
#pragma unroll
    for (int v = 0; v < 8; ++v) {
        sidx[v] = ei[rbase + v];
        didx[v] = ei[EE + rbase + v];
    }

    // Preload the wave's A fragments (reused by every head / sub-tile).
    const float* __restrict__ ap = edge_attr + (size_t)(wid * 16 + l16) * DD + 2 * hl;
    v2f afrag[DD / 4];
#pragma unroll
    for (int kb = 0; kb < DD / 4; ++kb)
        afrag[kb] = *(const v2f*)(ap + 4 * kb);

    for (int h = 0; h < HH; ++h) {
        float acc[8] = {0.f, 0.f, 0.f, 0.f, 0.f, 0.f, 0.f, 0.f};
#pragma unroll
        for (int sub = 0; sub < 2; ++sub) {
            const int gcol = h * CC + sub * 16 + l16;
            v8f c = {};
#pragma unroll
            for (int kb = 0; kb < DD / 4; ++kb) {
                const int k2 = 4 * kb + 2 * hl;
                v2f b;
                b[0] = We[(size_t)k2 * DD + gcol];
                b[1] = We[(size_t)(k2 + 1) * DD + gcol];
                c = wmma_f32(afrag[kb], b, c);
            }
            const float av = att[gcol];   // att[h][c], row-major [H,C]
#pragma unroll
            for (int v = 0; v < 8; ++v) {
                float m = c[v] + xl[(size_t)sidx[v] * DD + gcol]
                               + xr[(size_t)didx[v] * DD + gcol];
                float lr = m > 0.f ? m : m * NEG_SLOPE;
                float val = lr * av;
#pragma unroll
                for (int mm = 1; mm < 16; mm <<= 1)   // stays inside 16-lane half
                    val += __shfl_xor(val, mm, 32);
                acc[v] += val;
            }
        }
        if (l16 == 0) {
#pragma unroll
            for (int v = 0; v < 8; ++v)
                score[(size_t)(rbase + v) * HH + h] = acc[v];
        }
    }
}

// ---------------------------------------------------------------------------
// Kernel 3/4: segment max / segment sum over incoming edges per (dst, head).
// ---------------------------------------------------------------------------
__global__ __launch_bounds__(256) void seg_max(
    const float* __restrict__ score, const int* __restrict__ ei,
    unsigned* __restrict__ smaxkey)
{
    const int t = blockIdx.x * blockDim.x + threadIdx.x;
    if (t >= EE * HH) return;
    const int e = t / HH, h = t - e * HH;
    const int dst = ei[EE + e];
    atomicMax(&smaxkey[dst * HH + h], fkey(score[t]));
}

__global__ __launch_bounds__(256) void seg_sum(
    const float* __restrict__ score, const int* __restrict__ ei,
    const unsigned* __restrict__ smaxkey, float* __restrict__ ssum)
{
    const int t = blockIdx.x * blockDim.x + threadIdx.x;
    if (t >= EE * HH) return;
    const int e = t / HH, h = t - e * HH;
    const int dst = ei[EE + e];
    const float mx = fdecode(smaxkey[dst * HH + h]);
    atomicAdd(&ssum[dst * HH + h], __expf(score[t] - mx));
}

// ---------------------------------------------------------------------------
// Kernel 5: aggregation  attn[dst] += xl[src] * alpha.  One wave per edge;
// lanes 0..4 compute alpha per head, shuffle-broadcast, 5 atomic f32 adds/lane.
// ---------------------------------------------------------------------------
__global__ __launch_bounds__(256) void aggregate(
    const float* __restrict__ score, const int* __restrict__ ei,
    const unsigned* __restrict__ smaxkey, const float* __restrict__ ssum,
    const float* __restrict__ xl, float* __restrict__ attn)
{
    const int e    = (blockIdx.x * blockDim.x + threadIdx.x) >> 5;
    const int lane = threadIdx.x & 31;
    if (e >= EE) return;
    const int src = ei[e];
    const int dst = ei[EE + e];

    float a = 0.f;
    if (lane < HH) {
        const float s  = score[(size_t)e * HH + lane];
        const float mx = fdecode(smaxkey[dst * HH + lane]);
        const float sm = ssum[dst * HH + lane];
        a = __expf(s - mx) / (sm == 0.f ? 1.f : sm);
    }
#pragma unroll
    for (int k = 0; k < HH; ++k) {
        const float ak = __shfl(a, k, 32);
        const int col = k * CC + lane;
        atomicAdd(&attn[(size_t)dst * DD + col],
                  xl[(size_t)src * DD + col] * ak);
    }
}

// ---------------------------------------------------------------------------
// Kernel 6: fused epilogue per node row (one wave per node):
// h = LN(x + attn + bias);  z = LN(selu(h@W1+b1));  out = LN(h + z@W2 + b2)
// ---------------------------------------------------------------------------
__global__ __launch_bounds__(256) void epilogue(
    const float* __restrict__ x, const float* __restrict__ attn,
    const float* __restrict__ bias,
    const float* __restrict__ g1, const float* __restrict__ beta1,
    const float* __restrict__ W1, const float* __restrict__ b1,
    const float* __restrict__ gm, const float* __restrict__ bm,
    const float* __restrict__ W2, const float* __restrict__ b2,
    const float* __restrict__ g2, const float* __restrict__ beta2,
    float* __restrict__ out)
{
    const int row  = (blockIdx.x * blockDim.x + threadIdx.x) >> 5;
    const int lane = threadIdx.x & 31;
    if (row >= NN) return;

    const float* __restrict__ xp = x    + (size_t)row * DD;
    const float* __restrict__ ap = attn + (size_t)row * DD;

    // ---- h = LN(x + attn + bias) over 160 elements (5 per lane) ----
    float h[5];
    float s = 0.f;
#pragma unroll
    for (int v = 0; v < 5; ++v) {
        const int col = lane + 32 * v;
        h[v] = xp[col] + ap[col] + bias[col];
        s += h[v];
    }
    float mu = wave_sum(s) * (1.f / DD);
    float s2 = 0.f;
#pragma unroll
    for (int v = 0; v < 5; ++v) { float d = h[v] - mu; s2 += d * d; }
    float rs = rsqrtf(wave_sum(s2) * (1.f / DD) + EPSLN);
#pragma unroll
    for (int v = 0; v < 5; ++v) {
        const int col = lane + 32 * v;
        h[v] = (h[v] - mu) * rs * g1[col] + beta1[col];
    }

    // ---- z = h @ W1 + b1 (50 outs: zA = lane, zB = lane+32 for lane<18) ----
    float zA = b1[lane];
    float zB = (lane < MH - 32) ? b1[lane + 32] : 0.f;
#pragma unroll
    for (int v = 0; v < 5; ++v) {
        for (int l = 0; l < 32; ++l) {
            const float hk = __shfl(h[v], l, 32);
            const int k = v * 32 + l;
            zA += hk * W1[(size_t)k * MH + lane];
            if (lane < MH - 32) zB += hk * W1[(size_t)k * MH + lane + 32];
        }
    }

    // ---- SELU ----
    const float SELU_S = 1.0507009873554805f, SELU_A = 1.6732632423543772f;
    zA = SELU_S * (zA > 0.f ? zA : SELU_A * (__expf(zA) - 1.f));
    zB = SELU_S * (zB > 0.f ? zB : SELU_A * (__expf(zB) - 1.f));

    // ---- LN over 50 ----
    float zs = zA + ((lane < MH - 32) ? zB : 0.f);
    mu = wave_sum(zs) * (1.f / MH);
    float dA = zA - mu, dB = zB - mu;
    float zv = dA * dA + ((lane < MH - 32) ? dB * dB : 0.f);
    rs = rsqrtf(wave_sum(zv) * (1.f / MH) + EPSLN);
    float znA = dA * rs * gm[lane] + bm[lane];
    float znB = (lane < MH - 32) ? (dB * rs * gm[lane + 32] + bm[lane + 32]) : 0.f;

    // ---- y = z @ W2 + b2 (160 outs, 5 per lane) ----
    float y[5];
#pragma unroll
    for (int v = 0; v < 5; ++v) y[v] = b2[lane + 32 * v];
    for (int j = 0; j < MH; ++j) {
        const float zj = (j < 32) ? __shfl(znA, j, 32) : __shfl(znB, j - 32, 32);
#pragma unroll
        for (int v = 0; v < 5; ++v)
            y[v] += zj * W2[(size_t)j * DD + lane + 32 * v];
    }

    // ---- out = LN(h + y) ----
    float t[5];
    s = 0.f;
#pragma unroll
    for (int v = 0; v < 5; ++v) { t[v] = h[v] + y[v]; s += t[v]; }
    mu = wave_sum(s) * (1.f / DD);
    s2 = 0.f;
#pragma unroll
    for (int v = 0; v < 5; ++v) { float d = t[v] - mu; s2 += d * d; }
    rs = rsqrtf(wave_sum(s2) * (1.f / DD) + EPSLN);
    float* __restrict__ op = out + (size_t)row * DD;
#pragma unroll
    for (int v = 0; v < 5; ++v) {
        const int col = lane + 32 * v;
        op[col] = (t[v] - mu) * rs * g2[col] + beta2[col];
    }
}

// ---------------------------------------------------------------------------
extern "C" void kernel_launch(void* const* d_in, const int* in_sizes, int n_in,
                              void* d_out, int out_size, void* d_ws, size_t ws_size,
                              hipStream_t stream)
{
    const float* x         = (const float*)d_in[0];
    const int*   ei        = (const int*)  d_in[1];
    const float* edge_attr = (const float*)d_in[2];
    // d_in[3] = u (unused), d_in[4] = batch (unused)
    const float* Wl    = (const float*)d_in[5];
    const float* bl    = (const float*)d_in[6];
    const float* Wr    = (const float*)d_in[7];
    const float* br    = (const float*)d_in[8];
    const float* We    = (const float*)d_in[9];
    const float* att   = (const float*)d_in[10];
    const float* bias  = (const float*)d_in[11];
    const float* g1    = (const float*)d_in[12];
    const float* beta1 = (const float*)d_in[13];
    const float* W1    = (const float*)d_in[14];
    const float* b1    = (const float*)d_in[15];
    const float* gm    = (const float*)d_in[16];
    const float* bm    = (const float*)d_in[17];
    const float* W2    = (const float*)d_in[18];
    const float* b2    = (const float*)d_in[19];
    const float* g2    = (const float*)d_in[20];
    const float* beta2 = (const float*)d_in[21];
    float* out = (float*)d_out;

    // workspace layout (floats): xl | xr | score | smaxkey | ssum | attn
    float*    xl      = (float*)d_ws;
    float*    xr      = xl + (size_t)NN * DD;
    float*    score   = xr + (size_t)NN * DD;
    unsigned* smaxkey = (unsigned*)(score + (size_t)EE * HH);
    float*    ssum    = (float*)(smaxkey + (size_t)NN * HH);
    float*    attn    = ssum + (size_t)NN * HH;

    hipMemsetAsync(smaxkey, 0, (size_t)NN * HH * sizeof(unsigned), stream);
    hipMemsetAsync(ssum,    0, (size_t)NN * HH * sizeof(float),    stream);
    hipMemsetAsync(attn,    0, (size_t)NN * DD * sizeof(float),    stream);

    const int BLK = 256;

    { // node transform: (N/16)*(D/16) waves
        const int waves = (NN / 16) * (DD / 16);
        node_xform<<<(waves * 32 + BLK - 1) / BLK, BLK, 0, stream>>>(
            x, Wl, bl, Wr, br, xl, xr);
    }
    { // edge scores: E/16 waves
        const int waves = EE / 16;
        edge_score<<<(waves * 32 + BLK - 1) / BLK, BLK, 0, stream>>>(
            edge_attr, We, ei, xl, xr, att, score);
    }
    { // softmax max & sum: E*H threads
        const int th = EE * HH;
        seg_max<<<(th + BLK - 1) / BLK, BLK, 0, stream>>>(score, ei, smaxkey);
        seg_sum<<<(th + BLK - 1) / BLK, BLK, 0, stream>>>(score, ei, smaxkey, ssum);
    }
    { // aggregation: E waves
        const long long th = (long long)EE * 32;
        aggregate<<<(int)((th + BLK - 1) / BLK), BLK, 0, stream>>>(
            score, ei, smaxkey, ssum, xl, attn);
    }
    { // fused LN + MLP + LN epilogue: N waves
        const long long th = (long long)NN * 32;
        epilogue<<<(int)((th + BLK - 1) / BLK), BLK, 0, stream>>>(
            x, attn, bias, g1, beta1, W1, b1, gm, bm, W2, b2, g2, beta2, out);
    }
}